// implicit_best_responder_52029233823792
// MI455X (gfx1250) — compile-verified
//
#include <hip/hip_runtime.h>
#include <math.h>

#define N 8192
#define KSPLIT 4
#define KCHUNK (N / KSPLIT) /* 2048 */

typedef __attribute__((ext_vector_type(2))) float v2f;
typedef __attribute__((ext_vector_type(4))) float v4f;
typedef __attribute__((ext_vector_type(8))) float v8f;

// ---------- wave32 / 32-wave block reductions ----------
__device__ __forceinline__ float warp_red_sum(float v) {
#pragma unroll
  for (int off = 16; off > 0; off >>= 1) v += __shfl_xor(v, off, 32);
  return v;
}
__device__ __forceinline__ float warp_red_max(float v) {
#pragma unroll
  for (int off = 16; off > 0; off >>= 1) v = fmaxf(v, __shfl_xor(v, off, 32));
  return v;
}
// blockDim.x must be 1024 (32 waves); every thread returns the block result.
__device__ __forceinline__ float block_red_sum(float v, float* red) {
  v = warp_red_sum(v);
  __syncthreads();
  if ((threadIdx.x & 31) == 0) red[threadIdx.x >> 5] = v;
  __syncthreads();
  return warp_red_sum(red[threadIdx.x & 31]);
}
__device__ __forceinline__ float block_red_max(float v, float* red) {
  v = warp_red_max(v);
  __syncthreads();
  if ((threadIdx.x & 31) == 0) red[threadIdx.x >> 5] = v;
  __syncthreads();
  return warp_red_max(red[threadIdx.x & 31]);
}

// ---------- kernel 1: q = softmax(logits) ----------
__global__ __launch_bounds__(1024) void k_softmax_q(const float* __restrict__ x,
                                                    float* __restrict__ q) {
  __shared__ float red[32];
  const int tid = threadIdx.x;
  float v[8];
  float lmax = -INFINITY;
#pragma unroll
  for (int j = 0; j < 8; ++j) {
    v[j] = x[tid + j * 1024];
    lmax = fmaxf(lmax, v[j]);
  }
  const float M = block_red_max(lmax, red);
  float lsum = 0.f;
#pragma unroll
  for (int j = 0; j < 8; ++j) {
    v[j] = expf(v[j] - M);
    lsum += v[j];
  }
  const float S = block_red_sum(lsum, red);
#pragma unroll
  for (int j = 0; j < 8; ++j) q[tid + j * 1024] = v[j] / S;
}

// ---------- kernel 2: cpart[by] = P[:, kchunk] @ q[kchunk] via V_WMMA_F32_16X16X4_F32 ----------
// Each wave owns 16 rows. A tile (16x4 f32): lane l holds P[r0+(l&15), k+2*(l>>4)+{0,1}].
// B tile (4x16) = q[k..k+3] broadcast to all columns: same per-lane K indexing as A.
// D: vgpr v, lanes 0-15 -> M=v, lanes 16-31 -> M=8+v; all columns identical, lanes N==0 store.
__global__ __launch_bounds__(256) void k_matvec_wmma(const float* __restrict__ P,
                                                     const float* __restrict__ q,
                                                     float* __restrict__ cpart) {
  __shared__ __align__(16) float qs[KCHUNK];  // 8 KB chunk of q
  const int k0 = blockIdx.y * KCHUNK;
  for (int i = threadIdx.x; i < KCHUNK / 4; i += 256)
    ((v4f*)qs)[i] = ((const v4f*)(q + k0))[i];
  __syncthreads();

  const int lane = threadIdx.x & 31;
  const int wave = threadIdx.x >> 5;
  const int r0   = blockIdx.x * 128 + wave * 16;
  const int m    = lane & 15;
  const int kh   = (lane >> 4) << 1;  // 0 or 2

  const float* arow = P + (size_t)(r0 + m) * N + (size_t)k0 + kh;
  const float* brow = qs + kh;

  v8f acc = {0.f, 0.f, 0.f, 0.f, 0.f, 0.f, 0.f, 0.f};
#pragma unroll 8
  for (int k = 0; k < KCHUNK; k += 4) {
    v2f a = *(const v2f*)(arow + k);  // global_load_b64, streams P exactly once
    v2f b = *(const v2f*)(brow + k);  // ds_load_b64 from staged q chunk
    acc = __builtin_amdgcn_wmma_f32_16x16x4_f32(
        /*neg_a=*/false, a, /*neg_b=*/false, b,
        /*c_mod=*/(short)0, acc, /*reuse_a=*/false, /*reuse_b=*/false);
  }

  if (m == 0) {  // lanes 0 and 16 hold column N=0 of D
    float* dst = cpart + (size_t)blockIdx.y * N + r0 + ((lane >> 4) << 3);
#pragma unroll
    for (int i = 0; i < 8; ++i) dst[i] = acc[i];
  }
}

// ---------- kernel 3: 64 Adam steps + implicit-gradient finalization ----------
__global__ __launch_bounds__(1024) void k_adam(const float* __restrict__ x0,
                                               const float* __restrict__ cpart,
                                               const float* __restrict__ lam_p,
                                               const float* __restrict__ lr_p,
                                               const int* __restrict__ iters_p,
                                               float* __restrict__ out) {
  __shared__ float red[32];
  const int tid = threadIdx.x;
  const float lam = lam_p[0];
  const float lr  = lr_p[0];
  const int   T   = iters_p[0];
  const float b1 = 0.9f, b2 = 0.999f, eps = 1e-8f;

  float x[8], mm[8], vv[8], c[8];
#pragma unroll
  for (int j = 0; j < 8; ++j) {
    const int i = tid + j * 1024;
    x[j]  = x0[i];
    c[j]  = cpart[i] + cpart[N + i] + cpart[2 * N + i] + cpart[3 * N + i];  // fixed-order k-split sum
    mm[j] = 0.f;
    vv[j] = 0.f;
  }

  float b1t = 1.f, b2t = 1.f;
  for (int t = 0; t < T; ++t) {
    float lmax = -INFINITY;
#pragma unroll
    for (int j = 0; j < 8; ++j) lmax = fmaxf(lmax, x[j]);
    const float M = block_red_max(lmax, red);
    float e[8], lsum = 0.f;
#pragma unroll
    for (int j = 0; j < 8; ++j) {
      e[j] = expf(x[j] - M);
      lsum += e[j];
    }
    const float S = block_red_sum(lsum, red);
    float s[8], ldot = 0.f;
#pragma unroll
    for (int j = 0; j < 8; ++j) {
      s[j] = e[j] / S;
      ldot += s[j] * c[j];
    }
    const float p = block_red_sum(ldot, red);  // payoff at x

    b1t *= b1;
    b2t *= b2;
#pragma unroll
    for (int j = 0; j < 8; ++j) {
      const float g = -(s[j] * (c[j] - p)) + lam * x[j];  // grad of loss
      mm[j] = b1 * mm[j] + (1.f - b1) * g;
      vv[j] = b2 * vv[j] + (1.f - b2) * g * g;
      const float mh = mm[j] / (1.f - b1t);
      const float vh = vv[j] / (1.f - b2t);
      x[j] -= lr * mh / (sqrtf(vh) + eps);
    }
  }

  // g = softmax(x) * (c - p) / lam ; br = (x - g) + g ; r = softmax(br) . c
  float lmax = -INFINITY;
#pragma unroll
  for (int j = 0; j < 8; ++j) lmax = fmaxf(lmax, x[j]);
  const float M = block_red_max(lmax, red);
  float e[8], lsum = 0.f;
#pragma unroll
  for (int j = 0; j < 8; ++j) {
    e[j] = expf(x[j] - M);
    lsum += e[j];
  }
  const float S = block_red_sum(lsum, red);
  float s[8], ldot = 0.f;
#pragma unroll
  for (int j = 0; j < 8; ++j) {
    s[j] = e[j] / S;
    ldot += s[j] * c[j];
  }
  const float p = block_red_sum(ldot, red);

  float br[8];
#pragma unroll
  for (int j = 0; j < 8; ++j) {
    const float g = s[j] * (c[j] - p) / lam;
    br[j] = (x[j] - g) + g;  // as written in reference (not simplified to x)
    out[1 + tid + j * 1024] = br[j];
  }

  float lmax2 = -INFINITY;
#pragma unroll
  for (int j = 0; j < 8; ++j) lmax2 = fmaxf(lmax2, br[j]);
  const float M2 = block_red_max(lmax2, red);
  float e2[8], lsum2 = 0.f;
#pragma unroll
  for (int j = 0; j < 8; ++j) {
    e2[j] = expf(br[j] - M2);
    lsum2 += e2[j];
  }
  const float S2 = block_red_sum(lsum2, red);
  float ldot2 = 0.f;
#pragma unroll
  for (int j = 0; j < 8; ++j) ldot2 += (e2[j] / S2) * c[j];
  const float r = block_red_sum(ldot2, red);
  if (tid == 0) out[0] = r;
}

extern "C" void kernel_launch(void* const* d_in, const int* in_sizes, int n_in,
                              void* d_out, int out_size, void* d_ws, size_t ws_size,
                              hipStream_t stream) {
  const float* logits = (const float*)d_in[0];  // (8192,)
  const float* P      = (const float*)d_in[1];  // (8192, 8192)
  const float* x0     = (const float*)d_in[2];  // (8192,)
  const float* lam    = (const float*)d_in[3];  // scalar
  const float* lr     = (const float*)d_in[4];  // scalar
  const int*   iters  = (const int*)d_in[5];    // scalar

  float* out   = (float*)d_out;        // [r, br...] = 8193 floats
  float* q     = (float*)d_ws;         // 8192 floats
  float* cpart = q + N;                // KSPLIT * 8192 floats

  k_softmax_q<<<1, 1024, 0, stream>>>(logits, q);
  dim3 grid(N / 128, KSPLIT);
  k_matvec_wmma<<<grid, 256, 0, stream>>>(P, q, cpart);
  k_adam<<<1, 1024, 0, stream>>>(x0, cpart, lam, lr, iters, out);
}